// VoxelCenterSampler_76897094468376
// MI455X (gfx1250) — compile-verified
//
#include <hip/hip_runtime.h>
#include <stdint.h>

// ---- problem constants (mirrors reference) ----
#define NPTS      150000
#define NXg       1536
#define NYg       1664
#define NZg       128
#define NBATCH    4
// key space = 4*1536*1664*128 = 1,308,622,848 bits
#define NWORDS    40894464u          // key space / 32
#define BITMAP_BYTES (NWORDS * 4ull) // 163,577,856 B
#define WORDS_PER_BLK 4096u          // 256 threads * 16 words
#define NBLK      9984u              // NWORDS / WORDS_PER_BLK (exact)

// ------------------------------------------------------------------
// Kernel 1: mark presence bits.  Stages the point tile into LDS via the
// CDNA5 async global->LDS data mover (tracked by ASYNCcnt).  Each thread
// emits 27 shifted keys; the 3 z-shifts of an (x,y) column differ by +-1
// in cz, so their bits are merged into one atomic when they share a word.
// ------------------------------------------------------------------
__global__ __launch_bounds__(256) void k_mark(const float* __restrict__ pts,
                                              unsigned* __restrict__ bitmap,
                                              int n) {
  __shared__ float4 tile[256];
  int i = blockIdx.x * 256 + threadIdx.x;

  if (i < n) {
    // LDS byte address: low 32 bits of the generic pointer are the LDS offset
    unsigned lds_addr = (unsigned)(unsigned long long)(&tile[threadIdx.x]);
    unsigned long long gaddr = (unsigned long long)(const void*)(pts + 4ull * i);
    asm volatile("global_load_async_to_lds_b128 %0, %1, off"
                 :: "v"(lds_addr), "v"(gaddr) : "memory");
  }
  asm volatile("s_wait_asynccnt 0" ::: "memory");
  __syncthreads();

  if (i >= n) return;
  float4 p = tile[threadIdx.x];
  int   b  = (int)p.x;
  float px = p.y, py = p.z, pz = p.w;

  // z-shift cell indices are reused for every (sx,sy) pair
  int czv[3];
  #pragma unroll
  for (int sz = 0; sz < 3; ++sz) {
    float z = pz + (float)(sz - 1) * 0.05f;
    czv[sz] = (int)floorf((z - (-3.0f)) / 0.05f);
  }

  #pragma unroll
  for (int sx = -1; sx <= 1; ++sx) {
    float x = px + (float)sx * 0.05f;
    int cx = (int)floorf((x - 0.0f) / 0.05f);
    #pragma unroll
    for (int sy = -1; sy <= 1; ++sy) {
      float y = py + (float)sy * 0.05f;
      int cy = (int)floorf((y - (-40.0f)) / 0.05f);
      unsigned keyBase = ((unsigned)(b * NXg + cx) * NYg + (unsigned)cy) * NZg;
      unsigned k0 = keyBase + (unsigned)czv[0];
      unsigned k1 = keyBase + (unsigned)czv[1];
      unsigned k2 = keyBase + (unsigned)czv[2];
      unsigned wA = k0 >> 5, mA = 1u << (k0 & 31u);
      unsigned wB = k1 >> 5, mB = 1u << (k1 & 31u);
      unsigned wC = k2 >> 5, mC = 1u << (k2 & 31u);
      bool doB = true, doC = true;
      if (wB == wA) { mA |= mB; doB = false; }
      if (wC == wA) { mA |= mC; doC = false; }
      else if (doB && wC == wB) { mB |= mC; doC = false; }
      atomicOr(&bitmap[wA], mA);
      if (doB) atomicOr(&bitmap[wB], mB);
      if (doC) atomicOr(&bitmap[wC], mC);
    }
  }
}

// ------------------------------------------------------------------
// Kernel 2: per-block popcount sums over 4096-word chunks (L2-resident).
// Wave32 shuffle reduce, one barrier.
// ------------------------------------------------------------------
__global__ __launch_bounds__(256) void k_count(const unsigned* __restrict__ bitmap,
                                               unsigned* __restrict__ blockSums) {
  unsigned base = blockIdx.x * WORDS_PER_BLK + threadIdx.x * 16u;
  const uint4* wp = (const uint4*)(bitmap + base);
  unsigned s = 0;
  #pragma unroll
  for (int k = 0; k < 4; ++k) {
    uint4 v = wp[k];
    s += __popc(v.x) + __popc(v.y) + __popc(v.z) + __popc(v.w);
  }
  #pragma unroll
  for (int off = 16; off > 0; off >>= 1) s += __shfl_down(s, off, 32);

  __shared__ unsigned wsum[8];
  if ((threadIdx.x & 31u) == 0u) wsum[threadIdx.x >> 5] = s;
  __syncthreads();
  if (threadIdx.x == 0) {
    unsigned t = 0;
    #pragma unroll
    for (int w = 0; w < 8; ++w) t += wsum[w];
    blockSums[blockIdx.x] = t;
  }
}

// ------------------------------------------------------------------
// Kernel 3: exclusive scan of the 9984 block sums, one 1024-thread
// workgroup (32 waves): shuffle scan per wave + wave-0 scan of totals.
// ------------------------------------------------------------------
__global__ __launch_bounds__(1024) void k_scan(unsigned* __restrict__ sums, int n) {
  __shared__ unsigned wtot[32];
  __shared__ unsigned woff[32];
  unsigned lane = threadIdx.x & 31u;
  unsigned wid  = threadIdx.x >> 5;
  unsigned carry = 0;

  for (int base = 0; base < n; base += 1024) {
    int i = base + (int)threadIdx.x;
    unsigned v = (i < n) ? sums[i] : 0u;

    // inclusive scan within wave
    unsigned incl = v;
    #pragma unroll
    for (int off = 1; off < 32; off <<= 1) {
      unsigned t = __shfl_up(incl, off, 32);
      if (lane >= (unsigned)off) incl += t;
    }

    __syncthreads();                       // protect wtot/woff reuse
    if (lane == 31u) wtot[wid] = incl;
    __syncthreads();

    if (threadIdx.x < 32u) {
      unsigned t = wtot[threadIdx.x];
      #pragma unroll
      for (int off = 1; off < 32; off <<= 1) {
        unsigned u = __shfl_up(t, off, 32);
        if (threadIdx.x >= (unsigned)off) t += u;
      }
      woff[threadIdx.x] = t;               // inclusive wave-total scan
    }
    __syncthreads();

    unsigned blockIncl = (wid ? woff[wid - 1] : 0u) + incl;
    if (i < n) sums[i] = carry + blockIncl - v;   // exclusive
    carry += woff[31];
  }
}

// ------------------------------------------------------------------
// Kernel 4: walk set bits in ascending key order, decode voxel center,
// apply modulo mask, scatter float4 rows at the unique-sorted rank.
// Wave32 shuffle scan + tiny cross-wave combine (2 barriers total).
// ------------------------------------------------------------------
__global__ __launch_bounds__(256) void k_emit(const unsigned* __restrict__ bitmap,
                                              const unsigned* __restrict__ blockOffsets,
                                              float4* __restrict__ out) {
  unsigned base = blockIdx.x * WORDS_PER_BLK + threadIdx.x * 16u;
  unsigned words[16];
  const uint4* wp = (const uint4*)(bitmap + base);
  unsigned s = 0;
  #pragma unroll
  for (int k = 0; k < 4; ++k) {
    uint4 v = wp[k];
    words[4*k+0] = v.x; words[4*k+1] = v.y; words[4*k+2] = v.z; words[4*k+3] = v.w;
  }
  #pragma unroll
  for (int k = 0; k < 16; ++k) s += __popc(words[k]);

  unsigned lane = threadIdx.x & 31u;
  unsigned wid  = threadIdx.x >> 5;

  // inclusive scan within wave
  unsigned incl = s;
  #pragma unroll
  for (int off = 1; off < 32; off <<= 1) {
    unsigned t = __shfl_up(incl, off, 32);
    if (lane >= (unsigned)off) incl += t;
  }

  __shared__ unsigned wtot[8];
  __shared__ unsigned woff[8];
  if (lane == 31u) wtot[wid] = incl;
  __syncthreads();
  if (threadIdx.x == 0) {
    unsigned acc = 0;
    #pragma unroll
    for (int w = 0; w < 8; ++w) { unsigned t = wtot[w]; woff[w] = acc; acc += t; }
  }
  __syncthreads();

  unsigned rank = blockOffsets[blockIdx.x] + woff[wid] + incl - s;  // exclusive

  #pragma unroll
  for (int k = 0; k < 16; ++k) {
    unsigned w = words[k];
    unsigned keyBase = (base + (unsigned)k) * 32u;
    while (w) {
      int bit = __ffs(w) - 1;
      w &= w - 1u;
      unsigned key = keyBase + (unsigned)bit;
      unsigned iz = key & 127u;
      unsigned r  = key >> 7;
      unsigned iy = r % (unsigned)NYg;  r /= (unsigned)NYg;
      unsigned ix = r % (unsigned)NXg;
      unsigned bo = r / (unsigned)NXg;
      if (((iz & 1u) == 1u) & ((ix & 1u) == 0u) & ((iy & 1u) == 0u)) {
        float cx = ((float)ix + 0.5f) * 0.05f + 0.0f;
        float cy = ((float)iy + 0.5f) * 0.05f + (-40.0f);
        float cz = ((float)iz + 0.5f) * 0.05f + (-3.0f);
        out[rank] = make_float4((float)bo, cx, cy, cz);
      }
      ++rank;
    }
  }
}

// ------------------------------------------------------------------
extern "C" void kernel_launch(void* const* d_in, const int* in_sizes, int n_in,
                              void* d_out, int out_size, void* d_ws, size_t ws_size,
                              hipStream_t stream) {
  const float* pts = (const float*)d_in[0];
  int n = in_sizes[0] / 4;                       // 150000 points

  unsigned* bitmap    = (unsigned*)d_ws;
  unsigned* blockSums = (unsigned*)((char*)d_ws + BITMAP_BYTES);

  // zero the presence bitmap and the whole output (masked/tail rows stay 0)
  hipMemsetAsync(bitmap, 0, BITMAP_BYTES, stream);
  hipMemsetAsync(d_out, 0, (size_t)out_size * sizeof(float), stream);

  k_mark <<<(n + 255) / 256, 256, 0, stream>>>(pts, bitmap, n);
  k_count<<<NBLK, 256, 0, stream>>>(bitmap, blockSums);
  k_scan <<<1, 1024, 0, stream>>>(blockSums, (int)NBLK);
  k_emit <<<NBLK, 256, 0, stream>>>(bitmap, blockSums, (float4*)d_out);
}